// DiffeoMeshDeformer_901943132675
// MI455X (gfx1250) — compile-verified
//
#include <hip/hip_runtime.h>

typedef __attribute__((ext_vector_type(2))) float v2f;
typedef __attribute__((ext_vector_type(8))) float v8f;

#define NUM_STEPS 30

// ---------------------------------------------------------------------------
// Affine transform of one homogeneous point per lane via V_WMMA_F32_16X16X4_F32.
// D(16x16) = A(16x4) x B(4x16):
//   A rows 0..3  = affine rows (uniform); A layout: lanes0-15 VGPR0=K0,VGPR1=K1;
//                  lanes16-31 VGPR0=K2,VGPR1=K3 (ISA 7.12.2).
//   B col n      = homog coords of point (g*16+n); mirrored striping:
//                  VGPR0 lanes0-15=K0 row, lanes16-31=K2 row; VGPR1 = K1 / K3.
//   D VGPR r, lanes0-15 = row r, col=lane  -> rows 0..2 are the outputs.
// Two WMMAs cover the 32 points of a wave. Requires full EXEC (guaranteed by
// wave-granular guards: 140000 % 32 == 0).
// ---------------------------------------------------------------------------
__device__ __forceinline__ void affine_wmma_transform(
    const float* __restrict__ M,  // 16 floats, row-major 4x4 (uniform per wave)
    float& x, float& y, float& z) // (d,h,w) in, transformed out (w_homog = 1)
{
    const int lane = threadIdx.x & 31;
    const int half = lane >> 4;
    const int ln   = lane & 15;

    v2f a;
    if (ln < 4) {
        a.x = M[ln * 4 + (half ? 2 : 0)];
        a.y = M[ln * 4 + (half ? 3 : 1)];
    } else {
        a.x = 0.0f; a.y = 0.0f;
    }

    v8f d0, d1;
    {   // points 0..15 of the wave
        float tx = __shfl(x, ln);
        float ty = __shfl(y, ln);
        float tz = __shfl(z, ln);
        v2f bm; bm.x = half ? tz : tx; bm.y = half ? 1.0f : ty;
        v8f c = {};
        d0 = __builtin_amdgcn_wmma_f32_16x16x4_f32(false, a, false, bm,
                                                   (short)0, c, false, false);
    }
    {   // points 16..31 of the wave
        float tx = __shfl(x, ln + 16);
        float ty = __shfl(y, ln + 16);
        float tz = __shfl(z, ln + 16);
        v2f bm; bm.x = half ? tz : tx; bm.y = half ? 1.0f : ty;
        v8f c = {};
        d1 = __builtin_amdgcn_wmma_f32_16x16x4_f32(false, a, false, bm,
                                                   (short)0, c, false, false);
    }

    // Gather result coords back: point p reads D[r] from lane (p&15) of its group.
    float r0a = __shfl(d0[0], ln), r0b = __shfl(d1[0], ln);
    float r1a = __shfl(d0[1], ln), r1b = __shfl(d1[1], ln);
    float r2a = __shfl(d0[2], ln), r2b = __shfl(d1[2], ln);
    x = half ? r0b : r0a;
    y = half ? r1b : r1a;
    z = half ? r2b : r2a;
}

// ---------------------------------------------------------------------------
// 4x4 inverse (adjugate). One thread per batch. Column-major code applied to
// row-major data: inv(M^T)^T == inv(M), so layout cancels.
// ---------------------------------------------------------------------------
__global__ void inv4x4_kernel(const float* __restrict__ A,
                              float* __restrict__ Ainv, int B)
{
    int b = blockIdx.x * blockDim.x + threadIdx.x;
    if (b >= B) return;
    const float* m = A + b * 16;
    float inv[16];

    inv[0]  =  m[5]*m[10]*m[15] - m[5]*m[11]*m[14] - m[9]*m[6]*m[15] + m[9]*m[7]*m[14] + m[13]*m[6]*m[11] - m[13]*m[7]*m[10];
    inv[4]  = -m[4]*m[10]*m[15] + m[4]*m[11]*m[14] + m[8]*m[6]*m[15] - m[8]*m[7]*m[14] - m[12]*m[6]*m[11] + m[12]*m[7]*m[10];
    inv[8]  =  m[4]*m[9]*m[15]  - m[4]*m[11]*m[13] - m[8]*m[5]*m[15] + m[8]*m[7]*m[13] + m[12]*m[5]*m[11] - m[12]*m[7]*m[9];
    inv[12] = -m[4]*m[9]*m[14]  + m[4]*m[10]*m[13] + m[8]*m[5]*m[14] - m[8]*m[6]*m[13] - m[12]*m[5]*m[10] + m[12]*m[6]*m[9];
    inv[1]  = -m[1]*m[10]*m[15] + m[1]*m[11]*m[14] + m[9]*m[2]*m[15] - m[9]*m[3]*m[14] - m[13]*m[2]*m[11] + m[13]*m[3]*m[10];
    inv[5]  =  m[0]*m[10]*m[15] - m[0]*m[11]*m[14] - m[8]*m[2]*m[15] + m[8]*m[3]*m[14] + m[12]*m[2]*m[11] - m[12]*m[3]*m[10];
    inv[9]  = -m[0]*m[9]*m[15]  + m[0]*m[11]*m[13] + m[8]*m[1]*m[15] - m[8]*m[3]*m[13] - m[12]*m[1]*m[11] + m[12]*m[3]*m[9];
    inv[13] =  m[0]*m[9]*m[14]  - m[0]*m[10]*m[13] - m[8]*m[1]*m[14] + m[8]*m[2]*m[13] + m[12]*m[1]*m[10] - m[12]*m[2]*m[9];
    inv[2]  =  m[1]*m[6]*m[15]  - m[1]*m[7]*m[14]  - m[5]*m[2]*m[15] + m[5]*m[3]*m[14] + m[13]*m[2]*m[7]  - m[13]*m[3]*m[6];
    inv[6]  = -m[0]*m[6]*m[15]  + m[0]*m[7]*m[14]  + m[4]*m[2]*m[15] - m[4]*m[3]*m[14] - m[12]*m[2]*m[7]  + m[12]*m[3]*m[6];
    inv[10] =  m[0]*m[5]*m[15]  - m[0]*m[7]*m[13]  - m[4]*m[1]*m[15] + m[4]*m[3]*m[13] + m[12]*m[1]*m[7]  - m[12]*m[3]*m[5];
    inv[14] = -m[0]*m[5]*m[14]  + m[0]*m[6]*m[13]  + m[4]*m[1]*m[14] - m[4]*m[2]*m[13] - m[12]*m[1]*m[6]  + m[12]*m[2]*m[5];
    inv[3]  = -m[1]*m[6]*m[11]  + m[1]*m[7]*m[10]  + m[5]*m[2]*m[11] - m[5]*m[3]*m[10] - m[9]*m[2]*m[7]   + m[9]*m[3]*m[6];
    inv[7]  =  m[0]*m[6]*m[11]  - m[0]*m[7]*m[10]  - m[4]*m[2]*m[11] + m[4]*m[3]*m[10] + m[8]*m[2]*m[7]   - m[8]*m[3]*m[6];
    inv[11] = -m[0]*m[5]*m[11]  + m[0]*m[7]*m[9]   + m[4]*m[1]*m[11] - m[4]*m[3]*m[9]  - m[8]*m[1]*m[7]   + m[8]*m[3]*m[5];
    inv[15] =  m[0]*m[5]*m[10]  - m[0]*m[6]*m[9]   - m[4]*m[1]*m[10] + m[4]*m[2]*m[9]  + m[8]*m[1]*m[6]   - m[8]*m[2]*m[5];

    float det  = m[0]*inv[0] + m[1]*inv[4] + m[2]*inv[8] + m[3]*inv[12];
    float rdet = 1.0f / det;
    float* o = Ainv + b * 16;
    #pragma unroll
    for (int i = 0; i < 16; ++i) o[i] = inv[i] * rdet;
}

// ---------------------------------------------------------------------------
// Main kernel: one thread per point. Affine (WMMA) -> 30 Euler steps of
// trilinear gather -> flow_int store -> inverse affine (WMMA) -> pred store.
// ---------------------------------------------------------------------------
__global__ __launch_bounds__(256)
void deform_kernel(const float* __restrict__ verts,
                   const float* __restrict__ affine,
                   const float* __restrict__ invaff,
                   const float* __restrict__ flow,
                   float* __restrict__ out_pred,    // [B,N,3]
                   float* __restrict__ out_fi,      // [B,3,N]
                   int N, int D, int H, int W)
{
    const int b   = blockIdx.y;
    const int gid = blockIdx.x * blockDim.x + threadIdx.x;
    if (gid >= N) return;  // N % 32 == 0 -> cut at wave boundary, EXEC stays full

    const float* vp = verts + ((size_t)b * N + gid) * 3;
    float pd = vp[0], ph = vp[1], pw = vp[2];

    // pred = affine @ [v,1]
    affine_wmma_transform(affine + (size_t)b * 16, pd, ph, pw);
    const float od = pd, oh = ph, ow = pw;

    const size_t CH = (size_t)D * H * W;
    const float* f0 = flow + (size_t)b * 3 * CH;
    const float* f1 = f0 + CH;
    const float* f2 = f0 + 2 * CH;

    const float dmax = (float)(D - 1);
    const float hmax = (float)(H - 1);
    const float wmax = (float)(W - 1);
    const float scale = 1.0f / (float)NUM_STEPS;

    #pragma unroll 1
    for (int s = 0; s < NUM_STEPS; ++s) {
        float cd = fminf(fmaxf(pd, 0.0f), dmax);
        float ch = fminf(fmaxf(ph, 0.0f), hmax);
        float cw = fminf(fmaxf(pw, 0.0f), wmax);

        float fld = floorf(cd), flh = floorf(ch), flw = floorf(cw);
        int d0 = (int)fld, h0 = (int)flh, w0 = (int)flw;
        int d1 = min(d0 + 1, D - 1);
        int h1 = min(h0 + 1, H - 1);
        int w1 = min(w0 + 1, W - 1);
        float fd = cd - fld, fh = ch - flh, fw = cw - flw;

        int b00 = (d0 * H + h0) * W;
        int b01 = (d0 * H + h1) * W;
        int b10 = (d1 * H + h0) * W;
        int b11 = (d1 * H + h1) * W;
        int o000 = b00 + w0, o001 = b00 + w1;
        int o010 = b01 + w0, o011 = b01 + w1;
        int o100 = b10 + w0, o101 = b10 + w1;
        int o110 = b11 + w0, o111 = b11 + w1;

        auto tri = [&](const float* __restrict__ f) -> float {
            float c000 = f[o000], c001 = f[o001];
            float c010 = f[o010], c011 = f[o011];
            float c100 = f[o100], c101 = f[o101];
            float c110 = f[o110], c111 = f[o111];
            float c00 = c000 * (1.0f - fw) + c001 * fw;
            float c01 = c010 * (1.0f - fw) + c011 * fw;
            float c10 = c100 * (1.0f - fw) + c101 * fw;
            float c11 = c110 * (1.0f - fw) + c111 * fw;
            float c0 = c00 * (1.0f - fh) + c01 * fh;
            float c1 = c10 * (1.0f - fh) + c11 * fh;
            return c0 * (1.0f - fd) + c1 * fd;
        };

        float rd = tri(f0);
        float rh = tri(f1);
        float rw = tri(f2);

        pd += rd * scale;
        ph += rh * scale;
        pw += rw * scale;
    }

    // flow_int = aux - pred0, channel-major [B,3,N]
    const size_t fib = (size_t)b * 3 * N;
    out_fi[fib + 0 * (size_t)N + gid] = pd - od;
    out_fi[fib + 1 * (size_t)N + gid] = ph - oh;
    out_fi[fib + 2 * (size_t)N + gid] = pw - ow;

    // pred = inv(affine) @ [aux,1]
    float qd = pd, qh = ph, qw = pw;
    affine_wmma_transform(invaff + (size_t)b * 16, qd, qh, qw);

    float* op = out_pred + ((size_t)b * N + gid) * 3;
    op[0] = qd; op[1] = qh; op[2] = qw;
}

extern "C" void kernel_launch(void* const* d_in, const int* in_sizes, int n_in,
                              void* d_out, int out_size, void* d_ws, size_t ws_size,
                              hipStream_t stream) {
    const float* verts  = (const float*)d_in[0];  // [B,N,3]
    const float* affine = (const float*)d_in[1];  // [B,4,4]
    const float* flow   = (const float*)d_in[2];  // [B,3,D,H,W]

    const int B = in_sizes[1] / 16;
    const int N = in_sizes[0] / (B * 3);
    const long long vox = (long long)in_sizes[2] / (3LL * B);
    int D = 1;
    while ((long long)(D + 1) * (D + 1) * (D + 1) <= vox) ++D;  // cubic volume
    const int H = D, W = D;

    float* invaff   = (float*)d_ws;              // B*16 floats scratch
    float* out_pred = (float*)d_out;             // B*N*3
    float* out_fi   = out_pred + (size_t)B * N * 3;  // B*3*N

    inv4x4_kernel<<<1, 32, 0, stream>>>(affine, invaff, B);

    dim3 block(256);
    dim3 grid((N + 255) / 256, B);
    deform_kernel<<<grid, block, 0, stream>>>(verts, affine, invaff, flow,
                                              out_pred, out_fi, N, D, H, W);
}